// HGAT_30193620091072
// MI455X (gfx1250) — compile-verified
//
#include <hip/hip_runtime.h>
#include <math.h>

#define NNODES   50000
#define NEDGES   500000
#define IN_DIM   256
#define HIDDEN   128
#define HEADS    4
#define HC       (HIDDEN * HEADS)   // 512
#define NCLS     10
#define NGRAPHS  512
#define ETOT     (NEDGES + NNODES)  // 550000 (self-loops appended)
#define NEG_SLOPE 0.2f

#define KC    32          // K-chunk staged in LDS
#define BPAD  80          // padded LDS row stride (floats): 2*80 mod 64 = 32 -> conflict-free frags

typedef float v2f __attribute__((ext_vector_type(2)));
typedef float v8f __attribute__((ext_vector_type(8)));

// ---------------------------------------------------------------- utilities
__global__ void fill_kernel(float* __restrict__ p, float v, int n) {
    int i = blockIdx.x * blockDim.x + threadIdx.x;
    if (i < n) p[i] = v;
}

__device__ inline void atomicMaxF(float* addr, float val) {
    int* ai = (int*)addr;
    int old = __float_as_int(*addr);
    while (__int_as_float(old) < val) {
        int assumed = old;
        old = atomicCAS(ai, assumed, __float_as_int(val));
        if (old == assumed) break;
    }
}

// -------------------------------------------------------- fp32 WMMA GEMM
// C[M,N] = A[M,K] @ B[K,N], row-major fp32. N == 512, K%32 == 0.
// Block = 256 threads (8 waves) -> macro-tile 128 rows x 64 cols.
// B K-chunks staged global->LDS with GLOBAL_LOAD_ASYNC_TO_LDS_B128 (ASYNCcnt),
// each wave: 16x64 slab = 4 accumulators, 4 WMMAs per A-fragment.
__global__ void gemm_wmma_f32_lds(const float* __restrict__ A,
                                  const float* __restrict__ B,
                                  float* __restrict__ C,
                                  int M, int K, int N) {
    __shared__ float Bs[KC * BPAD];

    int lane = (int)threadIdx.x & 31;
    int wv   = (int)threadIdx.x >> 5;        // 0..7
    int colBlk = blockIdx.x & 7;             // N/64 == 8 column blocks
    int rowBlk = blockIdx.x >> 3;
    int rowBase = rowBlk * 128 + wv * 16;
    int colBase = colBlk * 64;

    int r  = lane & 15;                      // A row-in-strip / B col-in-frag
    int kp = (lane >> 4) << 1;               // 0 (lanes 0-15) or 2 (lanes 16-31)

    int arow = rowBase + r;                  // clamp OOB rows; stores are guarded
    if (arow >= M) arow = M - 1;
    const float* Arow = A + (size_t)arow * K;

    // cooperative async-fill indices: thread t moves 32 bytes of the B chunk
    int t   = (int)threadIdx.x;
    int lkk = t >> 3;                        // 0..31  (row of chunk)
    int lc  = (t & 7) << 3;                  // 0,8,...,56 (col of chunk)
    unsigned ldsOff = (unsigned)(uintptr_t)&Bs[lkk * BPAD + lc];

    v8f acc0 = {}, acc1 = {}, acc2 = {}, acc3 = {};

    for (int k = 0; k < K; k += KC) {
        unsigned long long g =
            (unsigned long long)(uintptr_t)(B + (size_t)(k + lkk) * N + colBase + lc);
        // async global -> LDS, 2 x 16B per lane; offset applies to BOTH addresses
        asm volatile("global_load_async_to_lds_b128 %0, %1, off\n\t"
                     "global_load_async_to_lds_b128 %0, %1, off offset:16"
                     :: "v"(ldsOff), "v"(g) : "memory");
        asm volatile("s_wait_asynccnt 0" ::: "memory");
        __syncthreads();

#pragma unroll
        for (int kk = 0; kk < KC; kk += 4) {
            v2f a;
            a.x = Arow[k + kk + kp];
            a.y = Arow[k + kk + kp + 1];
            const float* brow0 = &Bs[(kk + kp) * BPAD + (lane & 15)];
            const float* brow1 = brow0 + BPAD;
            v2f b0; b0.x = brow0[0];  b0.y = brow1[0];
            v2f b1; b1.x = brow0[16]; b1.y = brow1[16];
            v2f b2; b2.x = brow0[32]; b2.y = brow1[32];
            v2f b3; b3.x = brow0[48]; b3.y = brow1[48];
            acc0 = __builtin_amdgcn_wmma_f32_16x16x4_f32(false, a, false, b0, (short)0, acc0, false, false);
            acc1 = __builtin_amdgcn_wmma_f32_16x16x4_f32(false, a, false, b1, (short)0, acc1, false, false);
            acc2 = __builtin_amdgcn_wmma_f32_16x16x4_f32(false, a, false, b2, (short)0, acc2, false, false);
            acc3 = __builtin_amdgcn_wmma_f32_16x16x4_f32(false, a, false, b3, (short)0, acc3, false, false);
        }
        __syncthreads();
    }

    // 16x16 f32 D layout: VGPR i -> row = i + 8*(lane>=16), col = lane&15
    int rb = rowBase + ((lane >> 4) << 3);
    int c  = colBase + (lane & 15);
#pragma unroll
    for (int i = 0; i < 8; ++i) {
        int row = rb + i;
        if (row < M) {
            float* cp = C + (size_t)row * N + c;
            cp[0]  = acc0[i];
            cp[16] = acc1[i];
            cp[32] = acc2[i];
            cp[48] = acc3[i];
        }
    }
}

// ------------------------------------------- attention score projections
__global__ void alpha_kernel(const float* __restrict__ H,
                             const float* __restrict__ a_src,
                             const float* __restrict__ a_dst,
                             float* __restrict__ aS,
                             float* __restrict__ aD) {
    int i = blockIdx.x * blockDim.x + threadIdx.x;   // n*HEADS + h
    if (i >= NNODES * HEADS) return;
    int n = i / HEADS, h = i % HEADS;
    const float* hp  = H + (size_t)n * HC + h * HIDDEN;
    const float* asp = a_src + h * HIDDEN;
    const float* adp = a_dst + h * HIDDEN;
    float s = 0.f, d = 0.f;
    for (int c = 0; c < HIDDEN; ++c) { float v = hp[c]; s += v * asp[c]; d += v * adp[c]; }
    aS[i] = s;
    aD[i] = d;
}

__device__ inline void edge_nodes(const int* __restrict__ ei, int e, int& s, int& d) {
    if (e < NEDGES) { s = ei[e]; d = ei[NEDGES + e]; }
    else            { s = d = e - NEDGES; }            // appended self-loop
}

// pass 1: leaky-relu logits + segment max (over incoming edges of dst)
__global__ void edge_logits(const int* __restrict__ ei,
                            const float* __restrict__ aS,
                            const float* __restrict__ aD,
                            float* __restrict__ ebuf,
                            float* __restrict__ mx) {
    int i = blockIdx.x * blockDim.x + threadIdx.x;   // e*HEADS + h
    if (i >= ETOT * HEADS) return;
    int e = i >> 2, h = i & 3;
    int s, d; edge_nodes(ei, e, s, d);
    float v = aS[s * HEADS + h] + aD[d * HEADS + h];
    v = v > 0.f ? v : NEG_SLOPE * v;
    ebuf[i] = v;
    atomicMaxF(&mx[d * HEADS + h], v);
}

// pass 2: ex = exp(e - max[dst]); segment sum denominator
__global__ void edge_exp(const int* __restrict__ ei,
                         float* __restrict__ ebuf,
                         const float* __restrict__ mx,
                         float* __restrict__ dn) {
    int i = blockIdx.x * blockDim.x + threadIdx.x;
    if (i >= ETOT * HEADS) return;
    int e = i >> 2, h = i & 3;
    int s, d; edge_nodes(ei, e, s, d);
    float ex = __expf(ebuf[i] - mx[d * HEADS + h]);
    ebuf[i] = ex;
    atomicAdd(&dn[d * HEADS + h], ex);
}

// pass 3: AGG[dst] += alpha * H[src]  (one block per edge, 128 thr x 4 dims)
__global__ void edge_agg(const int* __restrict__ ei,
                         const float* __restrict__ ebuf,
                         const float* __restrict__ dn,
                         const float* __restrict__ H,
                         float* __restrict__ AGG) {
    int e = blockIdx.x;
    int s, d; edge_nodes(ei, e, s, d);
    int c0 = (int)threadIdx.x * 4;       // 0..508, never crosses a head boundary
    int h  = c0 >> 7;
    float alpha = ebuf[e * HEADS + h] / (dn[d * HEADS + h] + 1e-16f);
    const float4 hv = *(const float4*)(H + (size_t)s * HC + c0);
    float* out = AGG + (size_t)d * HC + c0;
    atomicAdd(out + 0, alpha * hv.x);
    atomicAdd(out + 1, alpha * hv.y);
    atomicAdd(out + 2, alpha * hv.z);
    atomicAdd(out + 3, alpha * hv.w);
}

__global__ void bias_relu(float* __restrict__ H, const float* __restrict__ b, int total) {
    int i = blockIdx.x * blockDim.x + threadIdx.x;
    if (i >= total) return;
    float v = H[i] + b[i & (HC - 1)];
    H[i] = v > 0.f ? v : 0.f;
}

// global mean pool (sums + counts)
__global__ void pool_kernel(const float* __restrict__ H, const int* __restrict__ batch,
                            float* __restrict__ sums, float* __restrict__ cnts) {
    int i = blockIdx.x * blockDim.x + threadIdx.x;   // n*HC + c
    if (i >= NNODES * HC) return;
    int n = i >> 9, c = i & (HC - 1);
    int g = batch[n];
    atomicAdd(&sums[(size_t)g * HC + c], H[i]);
    if (c == 0) atomicAdd(&cnts[g], 1.0f);
}

__global__ void classifier(const float* __restrict__ sums, const float* __restrict__ cnts,
                           const float* __restrict__ Wc, const float* __restrict__ bc,
                           float* __restrict__ out) {
    int i = blockIdx.x * blockDim.x + threadIdx.x;   // g*NCLS + cls
    if (i >= NGRAPHS * NCLS) return;
    int g = i / NCLS, cls = i % NCLS;
    float cnt = cnts[g];
    float inv = 1.f / (cnt > 1.f ? cnt : 1.f);
    float acc = bc[cls];
    const float* sg = sums + (size_t)g * HC;
    for (int k = 0; k < HC; ++k) acc += (sg[k] * inv) * Wc[k * NCLS + cls];
    out[i] = acc;
}

// ------------------------------------------------------------------ driver
extern "C" void kernel_launch(void* const* d_in, const int* in_sizes, int n_in,
                              void* d_out, int out_size, void* d_ws, size_t ws_size,
                              hipStream_t stream) {
    const float* x    = (const float*)d_in[0];
    const int*   ei   = (const int*)  d_in[1];   // [2, NEDGES]
    const int*   batch= (const int*)  d_in[2];
    const float* W1   = (const float*)d_in[3];
    const float* as1  = (const float*)d_in[4];
    const float* ad1  = (const float*)d_in[5];
    const float* b1   = (const float*)d_in[6];
    const float* W2   = (const float*)d_in[7];
    const float* as2  = (const float*)d_in[8];
    const float* ad2  = (const float*)d_in[9];
    const float* b2   = (const float*)d_in[10];
    const float* Wc   = (const float*)d_in[11];
    const float* bc   = (const float*)d_in[12];
    float* out = (float*)d_out;

    float* ws = (float*)d_ws;
    size_t off = 0;
    auto take = [&](size_t n) { float* p = ws + off; off += (n + 255) & ~(size_t)255; return p; };
    float* Hlin = take((size_t)NNODES * HC);   // pre-aggregation features
    float* AGG  = take((size_t)NNODES * HC);   // aggregated / activated features
    float* aS   = take((size_t)NNODES * HEADS);
    float* aD   = take((size_t)NNODES * HEADS);
    float* mx   = take((size_t)NNODES * HEADS);
    float* dn   = take((size_t)NNODES * HEADS);
    float* eb   = take((size_t)ETOT * HEADS);
    float* sums = take((size_t)NGRAPHS * HC);
    float* cnts = take((size_t)NGRAPHS);
    (void)ws_size; (void)in_sizes; (void)n_in; (void)out_size;

    const int nh  = NNODES * HEADS;
    const int tot = NNODES * HC;
    const int eh  = ETOT * HEADS;

    auto run_layer = [&](const float* inp, int K, const float* W,
                         const float* asv, const float* adv, const float* bv) {
        int rowBlks = (NNODES + 127) / 128;                    // 391
        gemm_wmma_f32_lds<<<rowBlks * 8, 256, 0, stream>>>(inp, W, Hlin, NNODES, K, HC);
        alpha_kernel<<<(nh + 255) / 256, 256, 0, stream>>>(Hlin, asv, adv, aS, aD);
        fill_kernel<<<(nh + 255) / 256, 256, 0, stream>>>(mx, -3.0e38f, nh);
        fill_kernel<<<(nh + 255) / 256, 256, 0, stream>>>(dn, 0.f, nh);
        fill_kernel<<<(tot + 255) / 256, 256, 0, stream>>>(AGG, 0.f, tot);
        edge_logits<<<(eh + 255) / 256, 256, 0, stream>>>(ei, aS, aD, eb, mx);
        edge_exp   <<<(eh + 255) / 256, 256, 0, stream>>>(ei, eb, mx, dn);
        edge_agg   <<<ETOT, 128, 0, stream>>>(ei, eb, dn, Hlin, AGG);
        bias_relu  <<<(tot + 255) / 256, 256, 0, stream>>>(AGG, bv, tot);
    };

    run_layer(x,   IN_DIM, W1, as1, ad1, b1);   // layer 1: AGG = relu(GAT1(x))
    run_layer(AGG, HC,     W2, as2, ad2, b2);   // layer 2 (AGG consumed by GEMM before being zeroed)

    fill_kernel<<<(NGRAPHS * HC + 255) / 256, 256, 0, stream>>>(sums, 0.f, NGRAPHS * HC);
    fill_kernel<<<(NGRAPHS + 255) / 256, 256, 0, stream>>>(cnts, 0.f, NGRAPHS);
    pool_kernel<<<(tot + 255) / 256, 256, 0, stream>>>(AGG, batch, sums, cnts);
    classifier <<<(NGRAPHS * NCLS + 255) / 256, 256, 0, stream>>>(sums, cnts, Wc, bc, out);
}